// TextureL1Loss_34153579938303
// MI455X (gfx1250) — compile-verified
//
#include <hip/hip_runtime.h>
#include <math.h>

// ---------------------------------------------------------------------------
// TextureL1Loss for MI455X (gfx1250, wave32)
//   pred,target: [2,1,128,256,256] f32 -> scalar f32
//   sigmoid -> avgpool3(3,1,1) -> sobel3d -> |magP-magT| mean
// Fully LDS-resident separable pipeline per 16^3 tile; W-axis conv done on the
// matrix pipe via V_WMMA_F32_16X16X4_F32 with a constant banded B matrix.
// ---------------------------------------------------------------------------

typedef __attribute__((ext_vector_type(2))) float v2f;
typedef __attribute__((ext_vector_type(8))) float v8f;

#define DDIM 128
#define HDIM 256
#define WDIM 256
// tiles: 2 * (128/16) * (256/16) * (256/16) = 4096
#define NTILES 4096

// LDS float offsets (overlaid stages; lifetimes verified, 36504 floats = 146016B)
#define OFF_RAW 0      // [20][20][20] = 8000   raw (sigmoid'd / zero padded)
#define OFF_A1  8000   // [18][20][20] = 7200   avg along d
#define OFF_A2  0      // [18][18][20] = 6480   avg along d,h   (raw dead)
#define OFF_AV  6480   // [18][18][18] = 5832   full avg, masked to volume
#define OFF_SM  12312  // [16][18][18] = 5184   [1,1,1]_d (AV)
#define OFF_DF  17496  // [16][18][18] = 5184   [1,0,-1]_d (AV)
#define OFF_X   22680  // [16][16][18] = 4608   smooth_d,[1,2,1]_h  -> gx pre-w
#define OFF_Y   27288  // [16][16][18] = 4608   smooth_d,[1,0,-1]_h -> gy pre-w
#define OFF_Z   31896  // [16][16][18] = 4608   diff_d,[1,1,1]_h    -> gz pre-w
#define LDS_FLOATS 36504

// Banded weight matrix entry: out[n] = sum_k in[k] * w3[k-n], K padded 18->20.
// 1/9 (sobel norm) folded here.
__device__ __forceinline__ float band_coef(int k, int n, float w0, float w1, float w2) {
  int t = k - n;
  if (k >= 18 || t < 0 || t > 2) return 0.0f;
  float w = (t == 0) ? w0 : ((t == 1) ? w1 : w2);
  return w * (1.0f / 9.0f);
}

__device__ __forceinline__ v8f wmma_f32_16x16x4(v2f a, v2f b, v8f c) {
  // 8 args: (neg_a, A, neg_b, B, c_mod, C, reuse_a, reuse_b)
  return __builtin_amdgcn_wmma_f32_16x16x4_f32(false, a, false, b, (short)0, c,
                                               false, false);
}

__global__ void __launch_bounds__(256)
texture_l1_tile_kernel(const float* __restrict__ pred,
                       const float* __restrict__ tgt,
                       float* __restrict__ partial) {
  extern __shared__ float smem[];
  const int tid  = threadIdx.x;
  const int lane = tid & 31;
  const int wid  = tid >> 5;

  // tile decode: bc in [0,2), td in [0,8), th,tw in [0,16)
  const int bid = blockIdx.x;
  const int bc  = bid >> 11;
  const int r   = bid & 2047;
  const int td  = r >> 8;
  const int r2  = r & 255;
  const int th  = r2 >> 4;
  const int tw  = r2 & 15;
  const int D0 = td * 16, H0 = th * 16, W0 = tw * 16;

  // Per-lane constant B fragments (derivative & smooth bands), 5 K-chunks of 4.
  // B 4x16 layout (wave32): VGPR0: K=c*4+{0|2}, VGPR1: K+1; N = lane&15.
  v2f bder[5], bsmo[5];
  {
    const int n  = lane & 15;
    const int kb = (lane >> 4) << 1;  // 0 (lanes 0-15) or 2 (lanes 16-31)
#pragma unroll
    for (int c = 0; c < 5; ++c) {
      int k0 = 4 * c + kb;
      bder[c].x = band_coef(k0,     n, 1.0f, 0.0f, -1.0f);
      bder[c].y = band_coef(k0 + 1, n, 1.0f, 0.0f, -1.0f);
      bsmo[c].x = band_coef(k0,     n, 1.0f, 2.0f,  1.0f);
      bsmo[c].y = band_coef(k0 + 1, n, 1.0f, 2.0f,  1.0f);
    }
  }

  float acc = 0.0f;
  float magP[2][8];  // pred magnitudes kept in registers across phases

  for (int phase = 0; phase < 2; ++phase) {
    const float* __restrict__ src = (phase == 0) ? pred : tgt;

    // ---- raw 20^3 tile load: sigmoid for pred, zero pad outside the volume
    for (int idx = tid; idx < 8000; idx += 256) {
      const int i = idx / 400, j = (idx / 20) % 20, k = idx % 20;
      const int d = D0 - 2 + i, h = H0 - 2 + j, w = W0 - 2 + k;
      float v = 0.0f;
      if ((unsigned)d < DDIM && (unsigned)h < HDIM && (unsigned)w < WDIM) {
        const size_t g = (((size_t)bc * DDIM + d) * HDIM + h) * WDIM + w;
        const float x = src[g];
        if (phase == 0) {
          v = 1.0f / (1.0f + __expf(-x));
          __builtin_prefetch(&tgt[g], 0, 0);  // warm L2 for the target phase
        } else {
          v = x;
        }
      }
      smem[OFF_RAW + idx] = v;
    }
    __syncthreads();

    // ---- avg pass d: A1[18][20][20]
    for (int idx = tid; idx < 7200; idx += 256) {
      const float v = smem[OFF_RAW + idx] + smem[OFF_RAW + idx + 400] +
                      smem[OFF_RAW + idx + 800];
      smem[OFF_A1 + idx] = v * (1.0f / 3.0f);
    }
    __syncthreads();

    // ---- avg pass h: A2[18][18][20]  (overwrites dead raw region)
    for (int idx = tid; idx < 6480; idx += 256) {
      const int i = idx / 360, jk = idx % 360;
      const int j = jk / 20, k = jk % 20;
      const int a1 = i * 400 + j * 20 + k;
      const float v = smem[OFF_A1 + a1] + smem[OFF_A1 + a1 + 20] +
                      smem[OFF_A1 + a1 + 40];
      smem[OFF_A2 + idx] = v * (1.0f / 3.0f);
    }
    __syncthreads();

    // ---- avg pass w + volume mask: AV[18][18][18]
    // (reference zero-pads the avg *output* before sobel -> mask here)
    for (int idx = tid; idx < 5832; idx += 256) {
      const int i = idx / 324, jk = idx % 324;
      const int j = jk / 18, k = jk % 18;
      const int a2 = i * 360 + j * 20 + k;
      float v = (smem[OFF_A2 + a2] + smem[OFF_A2 + a2 + 1] +
                 smem[OFF_A2 + a2 + 2]) * (1.0f / 3.0f);
      const int d = D0 - 1 + i, h = H0 - 1 + j, w = W0 - 1 + k;
      if (!((unsigned)d < DDIM && (unsigned)h < HDIM && (unsigned)w < WDIM))
        v = 0.0f;
      smem[OFF_AV + idx] = v;
    }
    __syncthreads();

    // ---- sobel d: SM = [1,1,1]_d(AV), DF = [1,0,-1]_d(AV) : [16][18][18]
    for (int idx = tid; idx < 5184; idx += 256) {
      const float v0 = smem[OFF_AV + idx];
      const float v1 = smem[OFF_AV + idx + 324];
      const float v2 = smem[OFF_AV + idx + 648];
      smem[OFF_SM + idx] = v0 + v1 + v2;
      smem[OFF_DF + idx] = v0 - v2;
    }
    __syncthreads();

    // ---- sobel h: X=[1,2,1]h(SM), Y=[1,0,-1]h(SM), Z=[1,1,1]h(DF) : [16][16][18]
    for (int idx = tid; idx < 4608; idx += 256) {
      const int a = idx / 288, yc = idx % 288;
      const int y = yc / 18, c = yc % 18;
      const int si = a * 324 + y * 18 + c;
      const float s0 = smem[OFF_SM + si];
      const float s1 = smem[OFF_SM + si + 18];
      const float s2 = smem[OFF_SM + si + 36];
      smem[OFF_X + idx] = s0 + 2.0f * s1 + s2;
      smem[OFF_Y + idx] = s0 - s2;
      const float d0 = smem[OFF_DF + si];
      const float d1 = smem[OFF_DF + si + 18];
      const float d2 = smem[OFF_DF + si + 36];
      smem[OFF_Z + idx] = d0 + d1 + d2;
    }
    __syncthreads();

    // ---- sobel w on the matrix pipe: per wave, per dz row-group,
    //      D(16x16) = A(16x20) x B(20x16 banded) via 5 chained WMMA f32 16x16x4.
    {
      const int m  = lane & 15;          // A-matrix row (h index)
      const int kb = (lane >> 4) << 1;   // A K-pair base: 0 or 2
#pragma unroll
      for (int it = 0; it < 2; ++it) {
        const int aa = wid + it * 8;     // dz of this row-group (uniform per wave)
        const int rowX = OFF_X + (aa * 16 + m) * 18;
        const int rowY = OFF_Y + (aa * 16 + m) * 18;
        const int rowZ = OFF_Z + (aa * 16 + m) * 18;
        v8f gx = {0.f, 0.f, 0.f, 0.f, 0.f, 0.f, 0.f, 0.f};
        v8f gy = {0.f, 0.f, 0.f, 0.f, 0.f, 0.f, 0.f, 0.f};
        v8f gz = {0.f, 0.f, 0.f, 0.f, 0.f, 0.f, 0.f, 0.f};
#pragma unroll
        for (int c = 0; c < 5; ++c) {
          const int k0 = 4 * c + kb;
          v2f ax, ay, az;
          ax.x = (k0 < 18)     ? smem[rowX + k0]     : 0.0f;
          ax.y = (k0 + 1 < 18) ? smem[rowX + k0 + 1] : 0.0f;
          ay.x = (k0 < 18)     ? smem[rowY + k0]     : 0.0f;
          ay.y = (k0 + 1 < 18) ? smem[rowY + k0 + 1] : 0.0f;
          az.x = (k0 < 18)     ? smem[rowZ + k0]     : 0.0f;
          az.y = (k0 + 1 < 18) ? smem[rowZ + k0 + 1] : 0.0f;
          gx = wmma_f32_16x16x4(ax, bder[c], gx);
          gy = wmma_f32_16x16x4(ay, bsmo[c], gy);
          gz = wmma_f32_16x16x4(az, bsmo[c], gz);
        }
#pragma unroll
        for (int q = 0; q < 8; ++q) {
          const float mg =
              sqrtf(gx[q] * gx[q] + gy[q] * gy[q] + gz[q] * gz[q] + 1e-4f);
          if (phase == 0) magP[it][q] = mg;
          else            acc += fabsf(magP[it][q] - mg);
        }
      }
    }
    __syncthreads();  // before raw region is reused by next phase
  }

  // ---- deterministic block reduction -> one partial per tile
  smem[tid] = acc;
  __syncthreads();
  for (int st = 128; st > 0; st >>= 1) {
    if (tid < st) smem[tid] += smem[tid + st];
    __syncthreads();
  }
  if (tid == 0) partial[blockIdx.x] = smem[0];
}

__global__ void __launch_bounds__(256)
texture_l1_reduce_kernel(const float* __restrict__ partial,
                         float* __restrict__ out) {
  __shared__ float r[256];
  const int tid = threadIdx.x;
  float a = 0.0f;
#pragma unroll
  for (int j = 0; j < 16; ++j) a += partial[tid + j * 256];  // 4096 partials
  r[tid] = a;
  __syncthreads();
  for (int st = 128; st > 0; st >>= 1) {
    if (tid < st) r[tid] += r[tid + st];
    __syncthreads();
  }
  if (tid == 0) out[0] = r[0] * (1.0f / 16777216.0f);  // mean over 2*128*256*256
}

extern "C" void kernel_launch(void* const* d_in, const int* in_sizes, int n_in,
                              void* d_out, int out_size, void* d_ws,
                              size_t ws_size, hipStream_t stream) {
  (void)in_sizes; (void)n_in; (void)out_size; (void)ws_size;
  const float* pred = (const float*)d_in[0];
  const float* tgt  = (const float*)d_in[1];
  float* out        = (float*)d_out;
  float* partial    = (float*)d_ws;  // 4096 floats of scratch

  const size_t shmem = (size_t)LDS_FLOATS * sizeof(float);  // 146016 B (<160KB -> 2 WG/WGP)
  (void)hipFuncSetAttribute((const void*)texture_l1_tile_kernel,
                            hipFuncAttributeMaxDynamicSharedMemorySize,
                            (int)shmem);

  texture_l1_tile_kernel<<<NTILES, 256, shmem, stream>>>(pred, tgt, partial);
  texture_l1_reduce_kernel<<<1, 256, 0, stream>>>(partial, out);
}